// GCNEncoder_52115133170207
// MI455X (gfx1250) — compile-verified
//
#include <hip/hip_runtime.h>

// CDNA5 / gfx1250: wave32, WMMA f32 16x16x4 for exact-fp32 GEMM.
typedef float v2f __attribute__((ext_vector_type(2)));
typedef float v8f __attribute__((ext_vector_type(8)));

#define HIDC 64  // hidden channels (both layers output 64)

// ---------------------------------------------------------------------------
// GEMM: H[n_nodes x 64] = X[n_nodes x K] @ W[K x 64], all fp32.
// One wave computes a 16x64 output tile via V_WMMA_F32_16X16X4_F32.
// A-tile (16xK) staged in LDS with +1 row pad to avoid bank conflicts.
//
// ISA fragment layouts (CDNA5 ISA):
//   A 16x4 f32 : lane L holds M = L%16; VGPR j holds K = 2*(L/16) + j
//   B 4x16 f32 : lane L holds N = L%16; VGPR j holds K = 2*(L/16) + j
//   C/D 16x16  : VGPR r, lane L -> M = r + 8*(L/16), N = L%16
// ---------------------------------------------------------------------------
template <int K>
__global__ __launch_bounds__(128) void gemm_wmma_f32(
    const float* __restrict__ X, const float* __restrict__ W,
    float* __restrict__ H, int n_nodes) {
  __shared__ float lds[4 * 16 * (K + 1)];
  const int lane = threadIdx.x & 31;
  const int wave = threadIdx.x >> 5;
  const int m0 = (blockIdx.x * 4 + wave) * 16;  // 16-row tile base
  float* tl = &lds[wave * 16 * (K + 1)];
  const bool active = (m0 < n_nodes);

  if (active) {
    // cooperative, coalesced load of the wave's 16xK A-tile into padded LDS
    for (int i = lane; i < 16 * K; i += 32) {
      int r = i / K, k = i - r * K;
      tl[r * (K + 1) + k] = X[(long long)(m0 + r) * K + k];
    }
  }
  __syncthreads();  // includes DScnt wait
  if (!active) return;

  const int half = lane >> 4;  // 0 = lanes 0..15, 1 = lanes 16..31
  const int lm = lane & 15;

  v8f acc[4] = {};  // four 16x16 N-tiles -> 16x64

  for (int k0 = 0; k0 < K; k0 += 4) {
    const int kk = k0 + 2 * half;
    v2f a;
    a.x = tl[lm * (K + 1) + kk];
    a.y = tl[lm * (K + 1) + kk + 1];
#pragma unroll
    for (int nt = 0; nt < 4; ++nt) {
      const int n = nt * 16 + lm;
      v2f b;
      b.x = W[kk * HIDC + n];        // W is 32KB max -> L2/WGP$ resident
      b.y = W[(kk + 1) * HIDC + n];
      acc[nt] = __builtin_amdgcn_wmma_f32_16x16x4_f32(
          /*neg_a=*/false, a, /*neg_b=*/false, b,
          /*c_mod=*/(short)0, acc[nt], /*reuse_a=*/false, /*reuse_b=*/false);
    }
  }

  // store D: VGPR r, lanes 0-15 -> row m0+r; lanes 16-31 -> row m0+r+8
#pragma unroll
  for (int nt = 0; nt < 4; ++nt) {
#pragma unroll
    for (int r = 0; r < 8; ++r) {
      const int row = m0 + r + 8 * half;
      H[(long long)row * HIDC + nt * 16 + lm] = acc[nt][r];
    }
  }
}

// ---------------------------------------------------------------------------
// Edge gather/scale/scatter: agg[dst] += ew * h[src]   (per 64-ch row)
// 16 lanes per edge, float4 per lane -> each edge's gather is contiguous 64B
// segments; h (25.6 MB) is L2-resident so gathers + atomics stay in L2.
// ---------------------------------------------------------------------------
__global__ void edge_scatter(const int* __restrict__ src,
                             const int* __restrict__ dst,
                             const float* __restrict__ ew,
                             const float* __restrict__ h,
                             float* __restrict__ agg, long long n_edges) {
  long long idx = (long long)blockIdx.x * blockDim.x + threadIdx.x;
  const long long total = n_edges * 16;
  if (idx >= total) return;
  const long long e = idx >> 4;
  const int c = (int)(idx & 15) * 4;
  const int s = src[e];
  const int d = dst[e];
  const float w = ew[e];
  const float4 hv = *(const float4*)(h + (long long)s * HIDC + c);
  float* ap = agg + (long long)d * HIDC + c;
  atomicAdd(ap + 0, w * hv.x);
  atomicAdd(ap + 1, w * hv.y);
  atomicAdd(ap + 2, w * hv.z);
  atomicAdd(ap + 3, w * hv.w);
}

// h_out = relu(agg + bias); agg re-zeroed in place for the next layer.
__global__ void bias_relu_rezero(float* __restrict__ agg,
                                 const float* __restrict__ b,
                                 float* __restrict__ hout, int n) {
  int i = blockIdx.x * blockDim.x + threadIdx.x;
  if (i >= n) return;
  float v = agg[i] + b[i & (HIDC - 1)];
  hout[i] = v > 0.f ? v : 0.f;
  agg[i] = 0.f;
}

// layer-2 epilogue fused with pooling accumulation.
__global__ void bias_relu_pool(const float* __restrict__ agg,
                               const float* __restrict__ b,
                               const int* __restrict__ batch,
                               float* __restrict__ sums,
                               float* __restrict__ counts, int n_nodes) {
  int i = blockIdx.x * blockDim.x + threadIdx.x;
  if (i >= n_nodes * HIDC) return;
  const int node = i >> 6;
  const int c = i & (HIDC - 1);
  float v = agg[i] + b[c];
  v = v > 0.f ? v : 0.f;
  const int g = batch[node];
  atomicAdd(&sums[g * HIDC + c], v);
  if (c == 0) atomicAdd(&counts[g], 1.0f);
}

__global__ void finalize_pool(const float* __restrict__ sums,
                              const float* __restrict__ counts,
                              float* __restrict__ out, int total) {
  int i = blockIdx.x * blockDim.x + threadIdx.x;
  if (i >= total) return;
  const float cnt = counts[i >> 6];
  out[i] = sums[i] / fmaxf(cnt, 1.0f);
}

// ---------------------------------------------------------------------------
// inputs: x, edge_index(2,E), edge_weight, batch, num_graphs, W1, b1, W2, b2
// ---------------------------------------------------------------------------
extern "C" void kernel_launch(void* const* d_in, const int* in_sizes, int n_in,
                              void* d_out, int out_size, void* d_ws,
                              size_t ws_size, hipStream_t stream) {
  const float* x = (const float*)d_in[0];
  const int* ei = (const int*)d_in[1];
  const float* ew = (const float*)d_in[2];
  const int* batch = (const int*)d_in[3];
  const float* W1 = (const float*)d_in[5];
  const float* b1 = (const float*)d_in[6];
  const float* W2 = (const float*)d_in[7];
  const float* b2 = (const float*)d_in[8];

  const int n_nodes = in_sizes[0] / 128;       // IN_CH = 128
  const long long n_edges = in_sizes[2];
  const int n_graphs = out_size / HIDC;

  const int* esrc = ei;
  const int* edst = ei + n_edges;

  // workspace layout (all fp32)
  float* hbuf = (float*)d_ws;                         // n_nodes*64  (h / h1)
  float* agg = hbuf + (size_t)n_nodes * HIDC;         // n_nodes*64  (scatter acc)
  float* h2 = agg + (size_t)n_nodes * HIDC;           // n_nodes*64  (gemm2 out)
  float* sums = h2 + (size_t)n_nodes * HIDC;          // n_graphs*64
  float* counts = sums + (size_t)n_graphs * HIDC;     // n_graphs

  // deterministic init every call (graph-capture-safe)
  hipMemsetAsync(agg, 0, (size_t)n_nodes * HIDC * sizeof(float), stream);
  hipMemsetAsync(sums, 0, ((size_t)n_graphs * HIDC + n_graphs) * sizeof(float),
                 stream);

  const int tiles = (n_nodes + 15) / 16;
  const int gblocks = (tiles + 3) / 4;
  const long long ework = n_edges * 16;
  const int eblocks = (int)((ework + 255) / 256);
  const int n64 = n_nodes * HIDC;
  const int nblocks = (n64 + 255) / 256;

  // layer 1
  gemm_wmma_f32<128><<<gblocks, 128, 0, stream>>>(x, W1, hbuf, n_nodes);
  edge_scatter<<<eblocks, 256, 0, stream>>>(esrc, edst, ew, hbuf, agg, n_edges);
  bias_relu_rezero<<<nblocks, 256, 0, stream>>>(agg, b1, hbuf, n64);

  // layer 2
  gemm_wmma_f32<64><<<gblocks, 128, 0, stream>>>(hbuf, W2, h2, n_nodes);
  edge_scatter<<<eblocks, 256, 0, stream>>>(esrc, edst, ew, h2, agg, n_edges);
  bias_relu_pool<<<nblocks, 256, 0, stream>>>(agg, b2, batch, sums, counts,
                                              n_nodes);

  // mean pool
  finalize_pool<<<(out_size + 255) / 256, 256, 0, stream>>>(
      sums, counts, (float*)d_out, out_size);
}